// Breadth_56341380989600
// MI455X (gfx1250) — compile-verified
//
#include <hip/hip_runtime.h>
#include <math.h>

#define N_NODES   50000
#define D         96
#define NEG_SLOPE 0.2f
// K-pair interleaved W layout in LDS: element (k, c) lives at (k>>1)*KP + c*2 + (k&1).
// KP mod 64 == 32 -> the two 16-lane halves of a wave hit disjoint bank halves on ds_load_b64.
#define KP        224

typedef float v2f __attribute__((ext_vector_type(2)));
typedef float v8f __attribute__((ext_vector_type(8)));

#define USE_ASYNC_STAGE 1

__device__ __forceinline__ float leaky(float v) { return v > 0.f ? v : NEG_SLOPE * v; }

// IEEE-correct float atomic max via int/uint ordering trick (no NaNs expected here).
__device__ __forceinline__ void atomicMaxF32(float* addr, float val) {
    if (val >= 0.f) atomicMax((int*)addr, __float_as_int(val));
    else            atomicMin((unsigned int*)addr, (unsigned int)__float_as_int(val));
}

// ---------------------------------------------------------------------------
// Kernel 1: h = x @ W via V_WMMA_F32_16X16X4_F32, fused a_src/a_dst dots.
// Block = 256 threads = 8 waves; wave w owns rows [blk*128 + w*16, +16).
// ---------------------------------------------------------------------------
__global__ __launch_bounds__(256)
void gat_gemm_wmma(const float* __restrict__ x, const float* __restrict__ Wm,
                   const float* __restrict__ attS, const float* __restrict__ attD,
                   float* __restrict__ h, float* __restrict__ aS, float* __restrict__ aD)
{
    __shared__ __align__(16) float Wl[(D / 2) * KP];
    const int tid = threadIdx.x;

    // Stage W (96x96) into LDS, K-pair interleaved, via async-to-LDS scatter.
#if USE_ASYNC_STAGE
    {
        const unsigned ldsBase = (unsigned)(unsigned long long)(void*)&Wl[0]; // low 32b = LDS addr
        for (int idx = tid; idx < D * D; idx += 256) {
            int k = idx / D, c = idx - k * D;
            unsigned dstByte = ldsBase + (unsigned)(((k >> 1) * KP + c * 2 + (k & 1)) * 4);
            unsigned long long srcAddr = (unsigned long long)(const void*)(Wm + idx);
            asm volatile("global_load_async_to_lds_b32 %0, %1, off"
                         :: "v"(dstByte), "v"(srcAddr) : "memory");
        }
        asm volatile("s_wait_asynccnt 0" ::: "memory");
    }
#else
    for (int idx = tid; idx < D * D; idx += 256) {
        int k = idx / D, c = idx - k * D;
        Wl[(k >> 1) * KP + c * 2 + (k & 1)] = Wm[idx];
    }
#endif
    __syncthreads();

    const int wave = tid >> 5;
    const int lane = tid & 31;
    const int n    = lane & 15;   // N column within a 16-wide tile / M row for A loads
    const int half = lane >> 4;   // K sub-pair selector per ISA operand layout
    const int rowBase = blockIdx.x * 128 + wave * 16;
    const int aRow  = rowBase + n;
    const int aRowC = aRow < N_NODES ? aRow : (N_NODES - 1);  // clamp: keep EXEC all-ones for WMMA

    float attSc[6], attDc[6];
#pragma unroll
    for (int t = 0; t < 6; ++t) { attSc[t] = attS[t * 16 + n]; attDc[t] = attD[t * 16 + n]; }

    v8f acc[6] = {};
    const float* xrow  = x + (size_t)aRowC * D;
    const float* bbase = Wl + half * KP + n * 2;   // lane's B fragment base (8B aligned)

    for (int k0 = 0; k0 < D; k0 += 4) {
        const int kA = k0 + 2 * half;
        v2f a = *(const v2f*)(xrow + kA);          // A 16x4 f32: V0=K{0,2}, V1=K{1,3}
        const float* brow = bbase + (k0 >> 1) * KP;
#pragma unroll
        for (int t = 0; t < 6; ++t) {
            v2f b = *(const v2f*)(brow + t * 32);  // single ds_load_b64, even-aligned pair
            acc[t] = __builtin_amdgcn_wmma_f32_16x16x4_f32(
                false, a, false, b, (short)0, acc[t], false, false);
        }
    }

    // Unpack accumulators: VGPR r holds M=r (lanes 0-15) / M=r+8 (lanes 16-31), N = t*16+n.
    float af[6][8];
#pragma unroll
    for (int t = 0; t < 6; ++t) {
        v8f v = acc[t];
#pragma unroll
        for (int r = 0; r < 8; ++r) af[t][r] = v[r];
    }

#pragma unroll
    for (int r = 0; r < 8; ++r) {
        const int row = rowBase + r + 8 * half;   // same row across the 16 lanes of a half
        if (row < N_NODES) {
            float* hrow = h + (size_t)row * D;
#pragma unroll
            for (int t = 0; t < 6; ++t) hrow[t * 16 + n] = af[t][r];
        }
        float ls = 0.f, ld = 0.f;
#pragma unroll
        for (int t = 0; t < 6; ++t) { ls += af[t][r] * attSc[t]; ld += af[t][r] * attDc[t]; }
#pragma unroll
        for (int off = 1; off < 16; off <<= 1) {  // xor<16 stays within the half
            ls += __shfl_xor(ls, off, 32);
            ld += __shfl_xor(ld, off, 32);
        }
        if (n == 0 && row < N_NODES) { aS[row] = ls; aD[row] = ld; }
    }
}

// ---------------------------------------------------------------------------
// Kernel 2: seed segment-max with the self-loop value (every node has one).
// ---------------------------------------------------------------------------
__global__ void gat_node_init(const float* __restrict__ aS, const float* __restrict__ aD,
                              float* __restrict__ m)
{
    int i = blockIdx.x * blockDim.x + threadIdx.x;
    if (i < N_NODES) m[i] = leaky(aS[i] + aD[i]);
}

// ---------------------------------------------------------------------------
// Kernel 3: per-edge atomic max into m[dst].
// ---------------------------------------------------------------------------
__global__ void gat_edge_max(const int* __restrict__ src, const int* __restrict__ dst,
                             const float* __restrict__ aS, const float* __restrict__ aD,
                             float* __restrict__ m, int E)
{
    int e = blockIdx.x * blockDim.x + threadIdx.x;
    if (e < E) {
        int s = src[e], d = dst[e];
        atomicMaxF32(m + d, leaky(aS[s] + aD[d]));
    }
}

// ---------------------------------------------------------------------------
// Kernel 4: self-loop contribution: denom[i] = w_self, acc[i,:] = w_self*h[i,:].
// ---------------------------------------------------------------------------
__global__ void gat_node_self(const float* __restrict__ h, const float* __restrict__ aS,
                              const float* __restrict__ aD, const float* __restrict__ m,
                              float* __restrict__ denom, float* __restrict__ acc)
{
    int idx = blockIdx.x * blockDim.x + threadIdx.x;
    if (idx < N_NODES * D) {
        int i = idx / D;
        int f = idx - i * D;
        float w = expf(leaky(aS[i] + aD[i]) - m[i]);
        acc[idx] = w * h[idx];
        if (f == 0) denom[i] = w;
    }
}

// ---------------------------------------------------------------------------
// Kernel 5: one wave per edge; lane covers features {lane, lane+32, lane+64}.
// Unnormalized aggregate: acc[dst,:] += w * h[src,:]; denom[dst] += w.
// ---------------------------------------------------------------------------
__global__ __launch_bounds__(256)
void gat_edge_agg(const int* __restrict__ src, const int* __restrict__ dst,
                  const float* __restrict__ h, const float* __restrict__ aS,
                  const float* __restrict__ aD, const float* __restrict__ m,
                  float* __restrict__ denom, float* __restrict__ acc, int E)
{
    int e = (blockIdx.x << 3) + (threadIdx.x >> 5);
    if (e >= E) return;
    int lane = threadIdx.x & 31;
    int s = src[e], d = dst[e];
    const float* hs = h + (size_t)s * D;
    __builtin_prefetch(hs, 0, 3);                 // global_prefetch_b8
    float w = expf(leaky(aS[s] + aD[d]) - m[d]);  // uniform across the wave
    if (lane == 0) atomicAdd(denom + d, w);
    float* ac = acc + (size_t)d * D;
#pragma unroll
    for (int t = 0; t < 3; ++t) {
        int f = lane + 32 * t;
        atomicAdd(ac + f, w * hs[f]);
    }
}

// ---------------------------------------------------------------------------
// Kernel 6: out = tanh(acc/denom + bias), in place on d_out.
// ---------------------------------------------------------------------------
__global__ void gat_finalize(float* __restrict__ acc, const float* __restrict__ denom,
                             const float* __restrict__ bias)
{
    int idx = blockIdx.x * blockDim.x + threadIdx.x;
    if (idx < N_NODES * D) {
        int i = idx / D;
        int f = idx - i * D;
        acc[idx] = tanhf(acc[idx] / (denom[i] + 1e-16f) + bias[f]);
    }
}

extern "C" void kernel_launch(void* const* d_in, const int* in_sizes, int n_in,
                              void* d_out, int out_size, void* d_ws, size_t ws_size,
                              hipStream_t stream)
{
    const float* x    = (const float*)d_in[0];
    const float* Wm   = (const float*)d_in[1];
    const float* attS = (const float*)d_in[2];
    const float* attD = (const float*)d_in[3];
    const float* bias = (const float*)d_in[4];
    const int*   ei   = (const int*)d_in[5];
    const int E = in_sizes[5] / 2;
    const int* src = ei;        // edge_index[0] = message source
    const int* dst = ei + E;    // edge_index[1] = aggregation target

    // Workspace: h[N*D], aS[N], aD[N], m[N], denom[N]  (~20 MB). acc aliases d_out.
    float* ws    = (float*)d_ws;
    float* h     = ws;  ws += (size_t)N_NODES * D;
    float* aS    = ws;  ws += N_NODES;
    float* aD    = ws;  ws += N_NODES;
    float* m     = ws;  ws += N_NODES;
    float* denom = ws;
    float* acc   = (float*)d_out;

    gat_gemm_wmma<<<(N_NODES + 127) / 128, 256, 0, stream>>>(x, Wm, attS, attD, h, aS, aD);
    gat_node_init<<<(N_NODES + 255) / 256, 256, 0, stream>>>(aS, aD, m);
    gat_edge_max<<<(E + 255) / 256, 256, 0, stream>>>(src, dst, aS, aD, m, E);
    gat_node_self<<<(N_NODES * D + 255) / 256, 256, 0, stream>>>(h, aS, aD, m, denom, acc);
    gat_edge_agg<<<(E + 7) / 8, 256, 0, stream>>>(src, dst, h, aS, aD, m, denom, acc, E);
    gat_finalize<<<(N_NODES * D + 255) / 256, 256, 0, stream>>>(acc, denom, bias);
}